// WarpingOutput_21328807592198
// MI455X (gfx1250) — compile-verified
//
#include <hip/hip_runtime.h>

// Problem constants (fixed by the harness: B=16, H=512, W=512, C=3)
#define BATCH 16
#define HGT   512
#define WID   512
#define CH    3
#define NPIX  (BATCH * HGT * WID)          // 4,194,304 pixels per warp field
#define OUTSZ (NPIX * CH)                  // 12,582,912 floats per output
#define PXT   4                            // pixels per thread (same row)
#define TPB   256                          // 8 wave32 waves per block
#define PXB   (PXT * TPB)                  // 1024 pixels per block
#define BLKF  (NPIX / PXB)                 // 4096 blocks per field

// 4-byte-aligned vector types: gather addresses are only DWORD-aligned
// (pixel stride = 12 B). Flow loads / output stores use 16B-aligned f32x4.
typedef float f32x2 __attribute__((ext_vector_type(2), aligned(4)));
typedef float f32x4 __attribute__((ext_vector_type(4), aligned(4)));
typedef float f32x4a16 __attribute__((ext_vector_type(4), aligned(16)));

__global__ __launch_bounds__(TPB) void warp4_kernel(
    const float* __restrict__ frame0, const float* __restrict__ frame1,
    const float* __restrict__ f01,    const float* __restrict__ f10,
    const float* __restrict__ ft0,    const float* __restrict__ ft1,
    float* __restrict__ out)
{
    const int blk   = blockIdx.x;
    const int field = blk >> 12;                         // 0..3 (which warp)
    const int p0    = ((blk & (BLKF - 1)) << 10) + (threadIdx.x << 2);

    // 4 consecutive pixels share y and b (p0 is 4-aligned, W=512)
    const int xb = p0 & (WID - 1);
    const int y  = (p0 >> 9) & (HGT - 1);
    const int b  = p0 >> 18;

    // Uniform (per-wave) selection of source frame + flow field.
    const float* src;
    const float* flow;
    switch (field) {
        case 0:  src = frame1; flow = f01; break;  // warp(frame_1, f_01)
        case 1:  src = frame0; flow = f10; break;  // warp(frame_0, f_10)
        case 2:  src = frame0; flow = ft0; break;  // warp(frame_0, f_t0)
        default: src = frame1; flow = ft1; break;  // warp(frame_1, f_t1)
    }

    // Streaming flow read: two aligned b128 non-temporal loads (TH_LOAD_NT)
    // so the read-once flow stream never evicts the frames from the 192MB L2.
    const float* fbase = flow + p0 * 2;
    __builtin_prefetch(fbase + 2 * PXB, 0, 0);   // global_prefetch_b8, next block
    f32x4a16 flA = __builtin_nontemporal_load((const f32x4a16*)fbase);
    f32x4a16 flB = __builtin_nontemporal_load((const f32x4a16*)(fbase + 4));

    const float fly[PXT] = {flA.x, flA.z, flB.x, flB.z};
    const float flx[PXT] = {flA.y, flA.w, flB.y, flB.w};

    const float yf = (float)y;
    const int rowstride = WID * CH;
    const float* img = src + (b * HGT) * (WID * CH);

    float o[PXT][CH];
#pragma unroll
    for (int i = 0; i < PXT; ++i) {
        // query = grid - flow; floor clipped to [0, size-2]; alpha to [0,1]
        float qy = yf - fly[i];
        float qx = (float)(xb + i) - flx[i];
        float fy = fminf(fmaxf(floorf(qy), 0.0f), (float)(HGT - 2));
        float fx = fminf(fmaxf(floorf(qx), 0.0f), (float)(WID - 2));
        float ay = fminf(fmaxf(qy - fy, 0.0f), 1.0f);
        float ax = fminf(fmaxf(qx - fx, 0.0f), 1.0f);
        int y0 = (int)fy;
        int x0 = (int)fx;

        // tl+tr = 6 contiguous floats; bl+br one row below: b128 + b64 each.
        const float* tp = img + y0 * rowstride + x0 * CH;
        const float* bp = tp + rowstride;
        f32x4 t03 = *(const f32x4*)tp;
        f32x2 t45 = *(const f32x2*)(tp + 4);
        f32x4 b03 = *(const f32x4*)bp;
        f32x2 b45 = *(const f32x2*)(bp + 4);

        float t[6]  = {t03.x, t03.y, t03.z, t03.w, t45.x, t45.y};
        float bo[6] = {b03.x, b03.y, b03.z, b03.w, b45.x, b45.y};
#pragma unroll
        for (int c = 0; c < CH; ++c) {
            float top = fmaf(ax, t[c + 3] - t[c],  t[c]);
            float bot = fmaf(ax, bo[c + 3] - bo[c], bo[c]);
            o[i][c] = fmaf(ay, bot - top, top);
        }
    }

    // Streaming output: three aligned b128 non-temporal stores (TH_STORE_NT).
    float* op = out + field * OUTSZ + p0 * CH;   // 48B per thread, 16B aligned
    f32x4a16 s0, s1, s2;
    s0.x = o[0][0]; s0.y = o[0][1]; s0.z = o[0][2]; s0.w = o[1][0];
    s1.x = o[1][1]; s1.y = o[1][2]; s1.z = o[2][0]; s1.w = o[2][1];
    s2.x = o[2][2]; s2.y = o[3][0]; s2.z = o[3][1]; s2.w = o[3][2];
    __builtin_nontemporal_store(s0, (f32x4a16*)op);
    __builtin_nontemporal_store(s1, (f32x4a16*)(op + 4));
    __builtin_nontemporal_store(s2, (f32x4a16*)(op + 8));
}

extern "C" void kernel_launch(void* const* d_in, const int* in_sizes, int n_in,
                              void* d_out, int out_size, void* d_ws, size_t ws_size,
                              hipStream_t stream) {
    (void)in_sizes; (void)n_in; (void)out_size; (void)d_ws; (void)ws_size;
    const float* frame0 = (const float*)d_in[0];
    const float* frame1 = (const float*)d_in[1];
    const float* f01    = (const float*)d_in[2];
    const float* f10    = (const float*)d_in[3];
    const float* ft0    = (const float*)d_in[4];
    const float* ft1    = (const float*)d_in[5];
    float* out = (float*)d_out;

    dim3 grid(4 * BLKF, 1, 1);   // 16384 blocks, 1D (field id in high bits)
    dim3 block(TPB, 1, 1);
    warp4_kernel<<<grid, block, 0, stream>>>(frame0, frame1, f01, f10, ft0, ft1, out);
}